// FeatureExtractionLayer_523986010290
// MI455X (gfx1250) — compile-verified
//
#include <hip/hip_runtime.h>
#include <stdint.h>

// Problem constants (from the reference): B=64, C=256, H=W=64, k=8
#define NB        64
#define NC        256
#define HW        4096                 // 64*64
#define CHW       (NC * HW)            // 1,048,576 elems per batch
#define K         8
#define OUTC      (NC + 3)             // 259
#define OUT_PER_B (K * OUTC)           // 2072

#define BPB        64                  // blocks per batch (phase 1)
#define EPB        (CHW / BPB)         // 16384 elems per block
#define TILE_F     1024                // floats per tile = 256 thr * 4
#define NT         (EPB / TILE_F)      // 16 tiles per block

// ---------------- CDNA5 async-copy primitives (ASYNCcnt path) ----------------

__device__ __forceinline__ void async_ld_b128(unsigned ldsOff, const float* gaddr) {
  // GLOBAL_LOAD_ASYNC_TO_LDS_B128: per-lane 16B  global -> LDS, tracked by ASYNCcnt.
  // Non-temporal: 256MB stream exceeds the 192MB L2 -> don't pollute.
  asm volatile("global_load_async_to_lds_b128 %0, %1, off th:TH_LOAD_NT"
               :: "v"(ldsOff), "v"(gaddr) : "memory");
}
__device__ __forceinline__ void wait_async0() {
  asm volatile("s_wait_asynccnt 0x0" ::: "memory");
}
__device__ __forceinline__ void wait_async1() {
  asm volatile("s_wait_asynccnt 0x1" ::: "memory");
}

__device__ __forceinline__ unsigned long long umax64(unsigned long long a,
                                                     unsigned long long b) {
  return a > b ? a : b;
}

// ---------------- Phase 1: per-block top-8 over a 16K-element slice ----------

__global__ __launch_bounds__(256)
void topk_phase1(const float* __restrict__ x, unsigned long long* __restrict__ ws) {
  __shared__ __align__(16) float tiles[2][TILE_F];     // 8 KB double buffer
  __shared__ unsigned long long cand[256 * K];         // 16 KB candidates
  __shared__ unsigned long long red[256];

  const int tid = threadIdx.x;
  const int blk = blockIdx.x & (BPB - 1);
  const int b   = blockIdx.x >> 6;                     // / BPB
  const float* base = x + (size_t)b * CHW + (size_t)blk * EPB;

  // register-resident sorted (descending) top-8 keys
  unsigned long long t8[K];
#pragma unroll
  for (int i = 0; i < K; ++i) t8[i] = 0ull;

  // LDS byte addresses of this lane's 16B slot in each buffer
  // (generic LDS pointer low 32 bits == offset within wave's LDS allocation)
  const unsigned lds0 = (unsigned)(uintptr_t)&tiles[0][0] + (unsigned)tid * 16u;
  const unsigned lds1 = (unsigned)(uintptr_t)&tiles[1][0] + (unsigned)tid * 16u;

  // prime the pipeline
  async_ld_b128(lds0, base + tid * 4);

  for (int t = 0; t < NT; ++t) {
    if (t + 1 < NT) {
      async_ld_b128((t & 1) ? lds0 : lds1, base + (t + 1) * TILE_F + tid * 4);
      wait_async1();                                   // wait for tile t only
    } else {
      wait_async0();
    }
    // each lane reads exactly the 16B its own async op wrote -> no barrier
    const float4 v = *reinterpret_cast<const float4*>(&tiles[t & 1][tid * 4]);
    const unsigned idx0 = (unsigned)(blk * EPB + t * TILE_F + tid * 4);
    const float vv[4] = {v.x, v.y, v.z, v.w};
#pragma unroll
    for (int j = 0; j < 4; ++j) {
      // non-negative fp32 -> bit pattern is order-preserving; ~idx breaks
      // ties toward the lower index (matches lax.top_k ordering)
      const unsigned long long key =
          ((unsigned long long)__float_as_uint(vv[j]) << 32) |
          (unsigned long long)(unsigned)~(idx0 + j);
      if (key > t8[K - 1]) {
        t8[K - 1] = key;
#pragma unroll
        for (int s = K - 1; s >= 1; --s) {
          const unsigned long long hi = umax64(t8[s - 1], t8[s]);
          const unsigned long long lo = t8[s - 1] ^ t8[s] ^ hi;
          t8[s - 1] = hi;
          t8[s]     = lo;
        }
      }
    }
  }

  // block-wide reduce: union of per-thread top-8 contains the block top-8
#pragma unroll
  for (int i = 0; i < K; ++i) cand[tid + 256 * i] = t8[i];
  __syncthreads();

  for (int r = 0; r < K; ++r) {
    unsigned long long m = 0ull;
#pragma unroll
    for (int i = 0; i < K; ++i) m = umax64(m, cand[tid + 256 * i]);
    red[tid] = m;
    __syncthreads();
    for (int s = 128; s > 0; s >>= 1) {
      if (tid < s) red[tid] = umax64(red[tid], red[tid + s]);
      __syncthreads();
    }
    const unsigned long long mx = red[0];
    if (tid == 0) ws[((size_t)b * BPB + blk) * K + r] = mx;
#pragma unroll
    for (int i = 0; i < K; ++i)
      if (cand[tid + 256 * i] == mx) cand[tid + 256 * i] = 0ull;   // keys unique
    __syncthreads();
  }
}

// ---------------- Phase 2: 512 candidates -> final top-8, emit output -------

__global__ __launch_bounds__(256)
void topk_phase2(const unsigned long long* __restrict__ ws, float* __restrict__ out) {
  __shared__ unsigned long long cand[BPB * K];         // 512
  __shared__ unsigned long long red[256];
  __shared__ unsigned long long fin[K];

  const int tid = threadIdx.x;
  const int b   = blockIdx.x;
  const unsigned long long* wsb = ws + (size_t)b * (BPB * K);

  cand[tid]       = wsb[tid];
  cand[tid + 256] = wsb[tid + 256];
  __syncthreads();

  for (int r = 0; r < K; ++r) {
    red[tid] = umax64(cand[tid], cand[tid + 256]);
    __syncthreads();
    for (int s = 128; s > 0; s >>= 1) {
      if (tid < s) red[tid] = umax64(red[tid], red[tid + s]);
      __syncthreads();
    }
    const unsigned long long mx = red[0];
    if (tid == 0) fin[r] = mx;
    if (cand[tid]       == mx) cand[tid]       = 0ull;
    if (cand[tid + 256] == mx) cand[tid + 256] = 0ull;
    __syncthreads();
  }

  // every output element written exactly once (no zero-then-scatter race)
  for (int e = tid; e < OUT_PER_B; e += 256) {
    const int j = e / OUTC;
    const int c = e - j * OUTC;
    const unsigned long long key = fin[j];
    const unsigned idx = ~(unsigned)key;               // index within batch
    const unsigned ch  = idx >> 12;                    // / HW
    const unsigned rem = idx & (HW - 1);
    const unsigned yy  = rem >> 6;
    const unsigned xx  = rem & 63u;
    float o;
    if (c < NC)            o = (c == (int)ch) ? 1.0f : 0.0f;
    else if (c == NC)      o = __uint_as_float((unsigned)(key >> 32));
    else if (c == NC + 1)  o = (float)xx * (1.0f / 63.0f);
    else                   o = (float)yy * (1.0f / 63.0f);
    out[(size_t)b * OUT_PER_B + e] = o;
  }
}

// ---------------- launch -----------------------------------------------------

extern "C" void kernel_launch(void* const* d_in, const int* in_sizes, int n_in,
                              void* d_out, int out_size, void* d_ws, size_t ws_size,
                              hipStream_t stream) {
  const float* x = (const float*)d_in[0];
  unsigned long long* ws = (unsigned long long*)d_ws;  // needs 64*512*8 = 256 KB
  float* out = (float*)d_out;

  topk_phase1<<<dim3(NB * BPB), dim3(256), 0, stream>>>(x, ws);
  topk_phase2<<<dim3(NB), dim3(256), 0, stream>>>(ws, out);
}